// Propagater_996432413628
// MI455X (gfx1250) — compile-verified
//
#include <hip/hip_runtime.h>

typedef __attribute__((ext_vector_type(2))) float v2f;
typedef __attribute__((ext_vector_type(4))) float v4f;
typedef __attribute__((ext_vector_type(8))) float v8f;

#define N_NODES 500000
#define DIM     128
#define MSGDIM  128
#define BATCH   4096
#define KNEI    20
#define ALPHA   0.01f
#define TAU     200.0f

// ---------------------------------------------------------------------------
// Kernel 1: bulk copy memory_cell / memory_hidden into the output buffers.
// Non-temporal B128 traffic: 512MB total per array pair, streams at HBM rate.
// ---------------------------------------------------------------------------
__global__ void copy_b128_nt(const v4f* __restrict__ src, v4f* __restrict__ dst,
                             long n4) {
    long i = (long)blockIdx.x * blockDim.x + threadIdx.x;
    long stride = (long)gridDim.x * blockDim.x;
    for (; i < n4; i += stride) {
        v4f v = __builtin_nontemporal_load(src + i);
        __builtin_nontemporal_store(v, dst + i);
    }
}

// ---------------------------------------------------------------------------
// Kernel 2: msg = unique_messages[B,M] @ W_s[M,D] in full f32 precision using
// V_WMMA_F32_16X16X4_F32. One wave computes one 16x16 tile; K-loop of 32.
//   A 16x4 layout : lanes 0-15 -> M=lane, {VGPR0,VGPR1} = K {k0,k0+1}
//                   lanes 16-31 -> M=lane-16, K {k0+2,k0+3}
//   B 4x16 layout : mirrored by column.
//   C/D layout    : VGPR r, lane l -> row = r + 8*(l>=16), col = l%16.
// ---------------------------------------------------------------------------
__global__ __launch_bounds__(32)
void msg_gemm_wmma(const float* __restrict__ A,   // [BATCH, MSGDIM]
                   const float* __restrict__ W,   // [MSGDIM, DIM]
                   float* __restrict__ Cout) {    // [BATCH, DIM]
    const int lane = threadIdx.x;      // 0..31, full wave, EXEC all ones
    const int n0 = blockIdx.x * 16;    // column tile (DIM/16 = 8)
    const int m0 = blockIdx.y * 16;    // row tile    (BATCH/16 = 256)

    const int half = lane >> 4;        // 0 or 1
    const int l16  = lane & 15;

    const float* arow = A + (size_t)(m0 + l16) * MSGDIM;  // row for this lane
    const int    bcol = n0 + l16;

    v8f acc = {};
    #pragma unroll 4
    for (int k0 = 0; k0 < MSGDIM; k0 += 4) {
        const int ka = k0 + 2 * half;                 // even -> 8B aligned
        v2f a = *(const v2f*)(arow + ka);             // A[m][ka], A[m][ka+1]
        v2f b;
        b.x = W[(size_t)ka * DIM + bcol];             // W[ka][n]
        b.y = W[(size_t)(ka + 1) * DIM + bcol];       // W[ka+1][n]
        acc = __builtin_amdgcn_wmma_f32_16x16x4_f32(
            /*neg_a=*/false, a, /*neg_b=*/false, b,
            /*c_mod=*/(short)0, acc, /*reuse_a=*/false, /*reuse_b=*/false);
    }

    #pragma unroll
    for (int r = 0; r < 8; ++r) {
        int row = m0 + r + 8 * half;
        Cout[(size_t)row * DIM + bcol] = acc[r];
    }
}

// ---------------------------------------------------------------------------
// Kernel 3: per-b attention + gated update + scatter.
// Block = 128 threads (4 wave32). Wave w owns neighbors k = 5w..5w+4 for the
// score phase (gather row staged in LDS, dot with src row, wave reduce).
// Then softmax over K=20, gate from time deltas, and the row scatter.
// ---------------------------------------------------------------------------
__global__ __launch_bounds__(128)
void update_scatter(const float* __restrict__ cell,     // [N, D]
                    const float* __restrict__ msg,      // [B, D]
                    const float* __restrict__ ts,       // [B]
                    const float* __restrict__ et,       // [B, K]
                    const int*   __restrict__ uid,      // [B]
                    const int*   __restrict__ nbr,      // [B, K]
                    float* __restrict__ out_cell,       // [N, D]
                    float* __restrict__ out_hidden) {   // [N, D]
    __shared__ float s_neigh[KNEI][DIM];   // 10 KB of 320 KB WGP LDS
    __shared__ float s_score[KNEI];
    __shared__ float s_coef[KNEI];

    const int b    = blockIdx.x;
    const int tid  = threadIdx.x;          // 0..127
    const int wave = tid >> 5;             // 0..3
    const int lane = tid & 31;

    const int u = uid[b];
    const float* srcrow = cell + (size_t)u * DIM;

    // ---- score phase: each wave handles 5 of the 20 neighbors ----
    #pragma unroll
    for (int kk = 0; kk < 5; ++kk) {
        const int k = wave * 5 + kk;
        const int n = nbr[b * KNEI + k];
        const float* nrow = cell + (size_t)n * DIM;
        float partial = 0.0f;
        #pragma unroll
        for (int j = 0; j < 4; ++j) {
            const int d = lane + 32 * j;   // coalesced 128B per pass
            const float nv = nrow[d];
            s_neigh[k][d] = nv;
            partial += nv * srcrow[d];
        }
        // wave32 reduction
        #pragma unroll
        for (int off = 16; off > 0; off >>= 1)
            partial += __shfl_xor(partial, off, 32);
        if (lane == 0) s_score[k] = partial;
    }
    __syncthreads();

    // ---- softmax over K (redundant per thread, K=20 is tiny) ----
    float mx = -3.402823466e38f;
    #pragma unroll
    for (int k = 0; k < KNEI; ++k) mx = fmaxf(mx, s_score[k]);
    float sum = 0.0f;
    #pragma unroll
    for (int k = 0; k < KNEI; ++k) sum += expf(s_score[k] - mx);
    const float inv = 1.0f / sum;

    if (tid < KNEI) {
        const float d = ts[b] - et[b * KNEI + tid];
        const float g = (d > 0.0f && d < TAU) ? expf(-ALPHA * d) : 0.0f;
        s_coef[tid] = g * expf(s_score[tid] - mx) * inv;   // gate * att
    }
    __syncthreads();

    // ---- gated update + scatter: thread tid owns column d = tid ----
    const float m = msg[(size_t)b * DIM + tid];
    #pragma unroll 4
    for (int k = 0; k < KNEI; ++k) {
        const int n = nbr[b * KNEI + k];
        const float cv = s_neigh[k][tid] + s_coef[k] * m;
        out_cell  [(size_t)n * DIM + tid] = cv;
        out_hidden[(size_t)n * DIM + tid] = tanhf(cv);
    }
}

// ---------------------------------------------------------------------------
extern "C" void kernel_launch(void* const* d_in, const int* in_sizes, int n_in,
                              void* d_out, int out_size, void* d_ws, size_t ws_size,
                              hipStream_t stream) {
    const float* memory_cell   = (const float*)d_in[0];   // [N, D]
    const float* memory_hidden = (const float*)d_in[1];   // [N, D]
    const float* unique_msgs   = (const float*)d_in[2];   // [B, M]
    const float* W_s           = (const float*)d_in[3];   // [M, D]
    const float* timestamps    = (const float*)d_in[4];   // [B]
    const float* edge_times    = (const float*)d_in[5];   // [B, K]
    const int*   unique_ids    = (const int*)d_in[6];     // [B]
    const int*   neighbors     = (const int*)d_in[7];     // [B, K]

    float* out_cell   = (float*)d_out;                          // [N, D]
    float* out_hidden = (float*)d_out + (size_t)N_NODES * DIM;  // [N, D]
    float* msg_ws     = (float*)d_ws;                           // [B, D] scratch

    const long n4 = (long)N_NODES * DIM / 4;   // float4 count per array

    // 1) stream copies (scatter later overwrites the touched rows)
    copy_b128_nt<<<4096, 256, 0, stream>>>((const v4f*)memory_cell,   (v4f*)out_cell,   n4);
    copy_b128_nt<<<4096, 256, 0, stream>>>((const v4f*)memory_hidden, (v4f*)out_hidden, n4);

    // 2) msg = unique_messages @ W_s via f32 WMMA (one wave per 16x16 tile)
    dim3 gemm_grid(DIM / 16, BATCH / 16);      // (8, 256)
    msg_gemm_wmma<<<gemm_grid, 32, 0, stream>>>(unique_msgs, W_s, msg_ws);

    // 3) attention + gated update + row scatter
    update_scatter<<<BATCH, DIM, 0, stream>>>(memory_cell, msg_ws, timestamps,
                                              edge_times, unique_ids, neighbors,
                                              out_cell, out_hidden);
}